// EGraphSAGE_8229157339892
// MI455X (gfx1250) — compile-verified
//
#include <hip/hip_runtime.h>

#define N_NODES 50000
#define N_EDGES 800000
#define FEAT    128

typedef float v2f __attribute__((ext_vector_type(2)));
typedef float v8f __attribute__((ext_vector_type(8)));

// ---------------------------------------------------------------------------
// Scatter edge features onto dst (node init) and src (edge_agg) sums + counts.
// One wave (32 lanes) per edge; each lane owns 4 contiguous floats.
// ---------------------------------------------------------------------------
__global__ void __launch_bounds__(256)
edge_scatter_kernel(const float* __restrict__ edge_attr,
                    const long long* __restrict__ src,
                    const long long* __restrict__ dst,
                    float* __restrict__ sum_dst,
                    float* __restrict__ sum_src,
                    float* __restrict__ cnt_dst,
                    float* __restrict__ cnt_src)
{
    const int e    = (blockIdx.x * blockDim.x + threadIdx.x) >> 5;
    const int lane = threadIdx.x & 31;
    if (e >= N_EDGES) return;
    const int s = (int)src[e];
    const int d = (int)dst[e];
    const float4 v = reinterpret_cast<const float4*>(edge_attr + (size_t)e * FEAT)[lane];
    float* pd = sum_dst + (size_t)d * FEAT + lane * 4;
    float* ps = sum_src + (size_t)s * FEAT + lane * 4;
    atomicAdd(pd + 0, v.x); atomicAdd(pd + 1, v.y);
    atomicAdd(pd + 2, v.z); atomicAdd(pd + 3, v.w);
    atomicAdd(ps + 0, v.x); atomicAdd(ps + 1, v.y);
    atomicAdd(ps + 2, v.z); atomicAdd(ps + 3, v.w);
    if (lane == 0) {
        atomicAdd(cnt_dst + d, 1.0f);
        atomicAdd(cnt_src + s, 1.0f);
    }
}

// counts -> 1/max(count,1), in place (cnt_dst and cnt_src are contiguous)
__global__ void inv_count_kernel(float* __restrict__ c, int n)
{
    const int i = blockIdx.x * blockDim.x + threadIdx.x;
    if (i < n) c[i] = 1.0f / fmaxf(c[i], 1.0f);
}

// ---------------------------------------------------------------------------
// Gather h[src[e]] row and atomically accumulate into neigh[dst[e]].
// ---------------------------------------------------------------------------
__global__ void __launch_bounds__(256)
gather_scatter_kernel(const float* __restrict__ h,
                      const long long* __restrict__ src,
                      const long long* __restrict__ dst,
                      float* __restrict__ neigh)
{
    const int e    = (blockIdx.x * blockDim.x + threadIdx.x) >> 5;
    const int lane = threadIdx.x & 31;
    if (e >= N_EDGES) return;
    const int s = (int)src[e];
    const int d = (int)dst[e];
    const float4 v = reinterpret_cast<const float4*>(h + (size_t)s * FEAT)[lane];
    float* pd = neigh + (size_t)d * FEAT + lane * 4;
    atomicAdd(pd + 0, v.x); atomicAdd(pd + 1, v.y);
    atomicAdd(pd + 2, v.z); atomicAdd(pd + 3, v.w);
}

// ---------------------------------------------------------------------------
// out = relu( [A1*s1 | A2*s2] @ W + bias ), A* are [N,128], W is [256,128].
// s1/s2 == nullptr means scale 1.0 (used when A is already a mean / layer out).
// One wave computes a 16x16 output tile via V_WMMA_F32_16X16X4_F32, K=256.
// Grid = 3125 blocks x 256 threads = 25000 waves = (50000/16)*(128/16) tiles.
// EXEC is all-ones everywhere (no guards needed: 50000 % 16 == 0).
// ---------------------------------------------------------------------------
__global__ void __launch_bounds__(256)
gemm_relu_wmma_kernel(const float* __restrict__ A1, const float* __restrict__ s1,
                      const float* __restrict__ A2, const float* __restrict__ s2,
                      const float* __restrict__ W,
                      const float* __restrict__ bias,
                      float* __restrict__ out)
{
    const int lane   = threadIdx.x & 31;
    const int waveId = threadIdx.x >> 5;
    const int tile   = blockIdx.x * 8 + waveId;   // 8 col-tiles per row band
    const int tileM  = (tile >> 3) << 4;
    const int tileN  = (tile & 7) << 4;

    // A fragment addressing (16x4 f32): lanes 0-15 -> K={0,1}, lanes 16-31 -> K={2,3}
    const int m    = lane & 15;
    const int kOff = (lane >> 4) << 1;
    const int rowA = tileM + m;
    const float rs1 = s1 ? s1[rowA] : 1.0f;
    const float rs2 = s2 ? s2[rowA] : 1.0f;
    const float* a1row = A1 + (size_t)rowA * FEAT;
    const float* a2row = A2 + (size_t)rowA * FEAT;
    // B fragment addressing (4x16 f32): mirrored layout with N on lanes
    const float* wcol  = W + tileN + (lane & 15);

    v8f acc = {};
    #pragma unroll 8
    for (int k = 0; k < 128; k += 4) {              // first half: A1 * s1
        v2f a, b;
        a.x = a1row[k + kOff]     * rs1;
        a.y = a1row[k + kOff + 1] * rs1;
        const float* wp = wcol + (size_t)(k + kOff) * FEAT;
        b.x = wp[0];
        b.y = wp[FEAT];
        acc = __builtin_amdgcn_wmma_f32_16x16x4_f32(false, a, false, b,
                                                    (short)0, acc, false, false);
    }
    #pragma unroll 8
    for (int k = 0; k < 128; k += 4) {              // second half: A2 * s2
        v2f a, b;
        a.x = a2row[k + kOff]     * rs2;
        a.y = a2row[k + kOff + 1] * rs2;
        const float* wp = wcol + (size_t)(128 + k + kOff) * FEAT;
        b.x = wp[0];
        b.y = wp[FEAT];
        acc = __builtin_amdgcn_wmma_f32_16x16x4_f32(false, a, false, b,
                                                    (short)0, acc, false, false);
    }

    // C/D layout: VGPR v -> M = v (lanes 0-15) / v+8 (lanes 16-31), N = lane%16
    const int rowD = tileM + ((lane >> 4) << 3);
    const int colD = tileN + (lane & 15);
    const float bv = bias[colD];
    #pragma unroll
    for (int v = 0; v < 8; ++v) {
        out[(size_t)(rowD + v) * FEAT + colD] = fmaxf(acc[v] + bv, 0.0f);
    }
}

// ---------------------------------------------------------------------------
// Final: edge_emb[e] = [node[src]|node[dst]], logits[e] = edge_emb[e].Wf + bf.
// One wave per edge; 8 floats per lane; shfl_xor tree for the GEMV dot.
// ---------------------------------------------------------------------------
__global__ void __launch_bounds__(256)
edge_output_kernel(const float* __restrict__ node,
                   const long long* __restrict__ src,
                   const long long* __restrict__ dst,
                   const float* __restrict__ Wf,
                   const float* __restrict__ bf,
                   float* __restrict__ logits,
                   float* __restrict__ edge_emb)
{
    const int e    = (blockIdx.x * blockDim.x + threadIdx.x) >> 5;
    const int lane = threadIdx.x & 31;
    if (e >= N_EDGES) return;
    const int s = (int)src[e];
    const int d = (int)dst[e];
    const float4 vs = reinterpret_cast<const float4*>(node + (size_t)s * FEAT)[lane];
    const float4 vd = reinterpret_cast<const float4*>(node + (size_t)d * FEAT)[lane];
    float4* eo = reinterpret_cast<float4*>(edge_emb + (size_t)e * (2 * FEAT));
    eo[lane]      = vs;
    eo[lane + 32] = vd;
    const float4 w0 = reinterpret_cast<const float4*>(Wf)[lane];
    const float4 w1 = reinterpret_cast<const float4*>(Wf)[lane + 32];
    float p = vs.x * w0.x + vs.y * w0.y + vs.z * w0.z + vs.w * w0.w
            + vd.x * w1.x + vd.y * w1.y + vd.z * w1.z + vd.w * w1.w;
    #pragma unroll
    for (int off = 16; off > 0; off >>= 1) p += __shfl_xor(p, off, 32);
    if (lane == 0) logits[e] = p + bf[0];
}

// ---------------------------------------------------------------------------
extern "C" void kernel_launch(void* const* d_in, const int* in_sizes, int n_in,
                              void* d_out, int out_size, void* d_ws, size_t ws_size,
                              hipStream_t stream)
{
    const float*     edge_attr = (const float*)d_in[0];
    const long long* ei        = (const long long*)d_in[1];   // int64 [2, E]
    const long long* src       = ei;
    const long long* dst       = ei + N_EDGES;
    const float* W0e = (const float*)d_in[2];  const float* b0e = (const float*)d_in[3];
    const float* W0n = (const float*)d_in[4];  const float* b0n = (const float*)d_in[5];
    const float* W1e = (const float*)d_in[6];  const float* b1e = (const float*)d_in[7];
    const float* W1n = (const float*)d_in[8];  const float* b1n = (const float*)d_in[9];
    const float* Wf  = (const float*)d_in[10]; const float* bf  = (const float*)d_in[11];

    const size_t NF = (size_t)N_NODES * FEAT;
    float* ws      = (float*)d_ws;
    float* sum_dst = ws;                       // [N,128] layer-0 node input sums
    float* sum_src = sum_dst + NF;             // [N,128] edge_agg sums (both layers)
    float* cnt_dst = sum_src + NF;             // [N] -> inv counts after inv_count
    float* cnt_src = cnt_dst + N_NODES;        // [N]
    float* h       = cnt_src + N_NODES;        // [N,128]
    float* neigh   = h + NF;                   // [N,128]
    float* nodeA   = neigh + NF;               // [N,128] node_attr after layer 0
    float* nodeB   = nodeA + NF;               // [N,128] node_attr after layer 1

    float* out_logits = (float*)d_out;                       // [E]
    float* out_eemb   = out_logits + N_EDGES;                // [E, 256]
    float* out_nemb   = out_eemb + (size_t)N_EDGES * 2 * FEAT; // [N, 128]

    const dim3 blk(256);
    const dim3 grdE((N_EDGES * 32 + 255) / 256);             // 1 wave / edge
    const dim3 grdG(N_NODES / 16);                           // 3125: 8 waves/blk
    const dim3 grdC((2 * N_NODES + 255) / 256);

    // zero sums + counts (contiguous region), then scatter + invert counts
    hipMemsetAsync(sum_dst, 0, (2 * NF + 2 * N_NODES) * sizeof(float), stream);
    edge_scatter_kernel<<<grdE, blk, 0, stream>>>(edge_attr, src, dst,
                                                  sum_dst, sum_src, cnt_dst, cnt_src);
    inv_count_kernel<<<grdC, blk, 0, stream>>>(cnt_dst, 2 * N_NODES);

    // ---- layer 0 ----
    gemm_relu_wmma_kernel<<<grdG, blk, 0, stream>>>(sum_dst, cnt_dst,
                                                    sum_src, cnt_src,
                                                    W0e, b0e, h);
    hipMemsetAsync(neigh, 0, NF * sizeof(float), stream);
    gather_scatter_kernel<<<grdE, blk, 0, stream>>>(h, src, dst, neigh);
    gemm_relu_wmma_kernel<<<grdG, blk, 0, stream>>>(h, (const float*)nullptr,
                                                    neigh, cnt_dst,
                                                    W0n, b0n, nodeA);

    // ---- layer 1 ----
    gemm_relu_wmma_kernel<<<grdG, blk, 0, stream>>>(nodeA, (const float*)nullptr,
                                                    sum_src, cnt_src,
                                                    W1e, b1e, h);
    hipMemsetAsync(neigh, 0, NF * sizeof(float), stream);
    gather_scatter_kernel<<<grdE, blk, 0, stream>>>(h, src, dst, neigh);
    gemm_relu_wmma_kernel<<<grdG, blk, 0, stream>>>(h, (const float*)nullptr,
                                                    neigh, cnt_dst,
                                                    W1n, b1n, nodeB);

    // ---- outputs ----
    edge_output_kernel<<<grdE, blk, 0, stream>>>(nodeB, src, dst, Wf, bf,
                                                 out_logits, out_eemb);
    hipMemcpyAsync(out_nemb, nodeB, NF * sizeof(float),
                   hipMemcpyDeviceToDevice, stream);
}